// LatentNet_45054206935510
// MI455X (gfx1250) — compile-verified
//
#include <hip/hip_runtime.h>
#include <hip/hip_bf16.h>
#include <math.h>

// ---------------------------------------------------------------------------
// LatentNet CP-decomposition evaluator for gfx1250 (MI455X).
//   Stage 1a: latent branches -> per-head constant vector cvec (3 tiny blocks)
//   Stage 1b: per-axis siren + head -> S matrices (384x64), diag folded into SU
//   Stage 2 : out[i,j,k] = sum_r SU'[i,r] * SV[j,r] * SW[k,r]
//             One wave fixes (i-tile, j), pre-scales A by the SV row in regs,
//             then sweeps all 24 k-tiles: 16 b64 loads + 16 V_WMMA_F32_16X16X4_F32
//             + 8 non-temporal stores per 16x16 tile. HBM traffic ~= the
//             226.5 MB output stream (9.7 us floor @ 23.3 TB/s).
// ---------------------------------------------------------------------------

#define Mdim 384
#define RANK 64
#define LATD 128
#define MIDD 256
#define OMG  30.0f

typedef __attribute__((ext_vector_type(2))) float v2f;
typedef __attribute__((ext_vector_type(8))) float v8f;

// ---------------- Stage 1a: cvec[o] = hb1[o] + sum_m L[m]*HW1[o, 256+m],
// where L = relu(lat @ LW1^T) @ LW2^T  (row-constant across the batch)
__global__ void latent_cvec_kernel(const float* __restrict__ lat,
                                   const float* __restrict__ LW1,   // (256,128)
                                   const float* __restrict__ LW2,   // (256,256)
                                   const float* __restrict__ HW1,   // (256,512)
                                   const float* __restrict__ hb1,   // (256,)
                                   float* __restrict__ cvec) {      // (256,)
    __shared__ float h1[MIDD];
    __shared__ float L[MIDD];
    const int m = threadIdx.x;
    float s = 0.f;
    for (int l = 0; l < LATD; ++l) s = fmaf(lat[l], LW1[m * LATD + l], s);
    h1[m] = fmaxf(s, 0.f);
    __syncthreads();
    float s2 = 0.f;
    for (int mm = 0; mm < MIDD; ++mm) s2 = fmaf(h1[mm], LW2[m * MIDD + mm], s2);
    L[m] = s2;
    __syncthreads();
    float c = hb1[m];
    for (int mm = 0; mm < MIDD; ++mm)
        c = fmaf(L[mm], HW1[m * (2 * MIDD) + MIDD + mm], c);
    cvec[m] = c;
}

// ---------------- Stage 1b: one block per row i; axis selected by args.
__global__ void axis_S_kernel(const float* __restrict__ XYZ, int axis,
                              const float* __restrict__ sW1,  // (256,1)
                              const float* __restrict__ sb1,  // (256,)
                              const float* __restrict__ sW2,  // (256,256)
                              const float* __restrict__ sb2,  // (256,)
                              const float* __restrict__ HW1,  // (256,512)
                              const float* __restrict__ cvec, // (256,)
                              const float* __restrict__ HW2,  // (64,256)
                              const float* __restrict__ hb2,  // (64,)
                              const float* __restrict__ CPdiag,
                              int foldDiag,
                              float* __restrict__ Sout) {     // (384,64)
    __shared__ float h1[MIDD];
    __shared__ float u[MIDD];
    __shared__ float hh[MIDD];
    const int i = blockIdx.x;
    const int o = threadIdx.x;
    const float x = XYZ[i * 3 + axis];

    h1[o] = sinf(OMG * fmaf(x, sW1[o], sb1[o]));
    __syncthreads();

    float s = sb2[o];
    for (int mm = 0; mm < MIDD; ++mm) s = fmaf(h1[mm], sW2[o * MIDD + mm], s);
    u[o] = sinf(OMG * s);
    __syncthreads();

    float s2 = cvec[o];
    for (int mm = 0; mm < MIDD; ++mm) s2 = fmaf(u[mm], HW1[o * (2 * MIDD) + mm], s2);
    hh[o] = sinf(OMG * s2);
    __syncthreads();

    if (o < RANK) {
        float s3 = hb2[o];
        for (int mm = 0; mm < MIDD; ++mm) s3 = fmaf(hh[mm], HW2[o * MIDD + mm], s3);
        if (foldDiag) s3 *= CPdiag[o];
        Sout[i * RANK + o] = s3;
    }
}

// ---------------- Stage 2: CP reconstruction via WMMA f32 16x16x4.
// Wave fixes (i-tile, j). A''[i,r] = SU'[i,r]*SV[j,r] is built once into 16
// register-resident fragments (A frag layout: lanes 0-15 -> M=lane,
// K={4t,4t+1}; lanes 16-31 -> M=lane-16, K={4t+2,4t+3}).  B frag = raw SW
// tile, same K striping with N across lanes.  Sweep kt=0..23, each tile:
// 16 b64 loads (L0-shared across the 8 waves of the block, which all have the
// same j) + 16 WMMAs + 8 NT stores of the C rows (VGPR v -> M=v / v+8).
__global__ void __launch_bounds__(256) cp_einsum_wmma(
        const float* __restrict__ SUp,   // (384,64) diag-folded
        const float* __restrict__ SVm,   // (384,64)
        const float* __restrict__ SWm,   // (384,64)
        float* __restrict__ out) {       // (384,384,384)
    const int lane = threadIdx.x & 31;
    const int wv   = threadIdx.x >> 5;
    const int w    = blockIdx.x * 8 + wv;     // 9216 waves total
    const int it   = w % 24;                  // waves in a block share j
    const int j    = w / 24;
    const int i0   = it * 16;
    const int col  = lane & 15;
    const int hi   = lane >> 4;               // K sub-offset: 0 -> +0, 1 -> +2

    const float* Ap = SUp + (i0 + col) * RANK + 2 * hi;
    const float* Vp = SVm + j * RANK + 2 * hi;

    // Pre-scaled A fragments, kept in VGPRs for all 24 k-tiles.
    v2f a[16];
#pragma unroll
    for (int t = 0; t < 16; ++t) {
        a[t].x = Ap[4 * t]     * Vp[4 * t];
        a[t].y = Ap[4 * t + 1] * Vp[4 * t + 1];
    }

    const size_t plane = (size_t)Mdim * Mdim;
    float* obase = out + (size_t)(i0 + 8 * hi) * plane + (size_t)j * Mdim + col;
    const float* Wp = SWm + col * RANK + 2 * hi;

    for (int kt = 0; kt < 24; ++kt) {
        const float* Wk = Wp + kt * 16 * RANK;
        v8f c = {};
#pragma unroll
        for (int t = 0; t < 16; ++t) {
            v2f b;
            b.x = Wk[4 * t];
            b.y = Wk[4 * t + 1];
            // D = A*B + C ; (neg_a, A, neg_b, B, c_mod, C, reuse_a, reuse_b)
            c = __builtin_amdgcn_wmma_f32_16x16x4_f32(
                    false, a[t], false, b, (short)0, c, false, false);
        }
        float* op = obase + kt * 16;
#pragma unroll
        for (int v = 0; v < 8; ++v)
            __builtin_nontemporal_store(c[v], op + (size_t)v * plane);
    }
}

// ---------------------------------------------------------------------------
// Input order: setup_inputs() dict insertion order, recursively flattened:
//  0 XYZaxis | 1 latent_x 2 latent_y 3 latent_z | 4 CPdiag
//  5 LU_W1 6 LU_W2 | 7 LV_W1 8 LV_W2 | 9 LW_W1 10 LW_W2
//  11..14 U{W1,b1,W2,b2} 15..18 V{...} 19..22 W{...}
//  23..26 SU{W1,b1,W2,b2} 27..30 SV{...} 31..34 SW{...}
// ---------------------------------------------------------------------------
extern "C" void kernel_launch(void* const* d_in, const int* in_sizes, int n_in,
                              void* d_out, int out_size, void* d_ws, size_t ws_size,
                              hipStream_t stream) {
    (void)in_sizes; (void)n_in; (void)out_size; (void)ws_size;
    const float* XYZ    = (const float*)d_in[0];
    const float* CPdiag = (const float*)d_in[4];

    float* ws   = (float*)d_ws;
    float* cvec = ws;                                  // 3*256 floats
    float* S0   = ws + 3 * MIDD;                       // SU' (diag folded)
    float* S1   = S0 + Mdim * RANK;                    // SV
    float* S2   = S1 + Mdim * RANK;                    // SW
    float* Sarr[3] = { S0, S1, S2 };

    for (int b = 0; b < 3; ++b) {
        latent_cvec_kernel<<<1, MIDD, 0, stream>>>(
            (const float*)d_in[1 + b],        // latent_{x,y,z}
            (const float*)d_in[5 + 2 * b],    // L?_W1
            (const float*)d_in[6 + 2 * b],    // L?_W2
            (const float*)d_in[23 + 4 * b],   // head W1
            (const float*)d_in[24 + 4 * b],   // head b1
            cvec + MIDD * b);
    }
    for (int b = 0; b < 3; ++b) {
        axis_S_kernel<<<Mdim, MIDD, 0, stream>>>(
            XYZ, b,
            (const float*)d_in[11 + 4 * b],   // siren W1
            (const float*)d_in[12 + 4 * b],   // siren b1
            (const float*)d_in[13 + 4 * b],   // siren W2
            (const float*)d_in[14 + 4 * b],   // siren b2
            (const float*)d_in[23 + 4 * b],   // head W1
            cvec + MIDD * b,
            (const float*)d_in[25 + 4 * b],   // head W2
            (const float*)d_in[26 + 4 * b],   // head b2
            CPdiag, (b == 0) ? 1 : 0,
            Sarr[b]);
    }
    // 24 i-tiles * 384 j = 9216 waves / 8 waves per block; each sweeps 24 k-tiles
    cp_einsum_wmma<<<1152, 256, 0, stream>>>(S0, S1, S2, (float*)d_out);
}